// GINNetwork_50921132261407
// MI455X (gfx1250) — compile-verified
//
#include <hip/hip_runtime.h>
#include <hip/hip_bf16.h>

typedef __attribute__((ext_vector_type(2))) float v2f;
typedef __attribute__((ext_vector_type(8))) float v8f;

#define HDIM 128
#define MTILES 4            // 16-row tiles per block => 64 rows/block

// ---------------------------------------------------------------------------
// Zero-fill scratch
// ---------------------------------------------------------------------------
__global__ void zero_kernel(float* __restrict__ p, size_t n) {
    size_t i = (size_t)blockIdx.x * blockDim.x + threadIdx.x;
    if (i < n) p[i] = 0.0f;
}

// ---------------------------------------------------------------------------
// Edge scatter-add:  AGG[dst] += X[src]   (float atomics, L2-resident target)
// One thread handles 4 features of one edge (float4 load, 4 atomic adds).
// ---------------------------------------------------------------------------
__global__ void scatter_add_kernel(const float* __restrict__ X,
                                   const int* __restrict__ src,
                                   const int* __restrict__ dst,
                                   float* __restrict__ AGG, int E) {
    long long gid = (long long)blockIdx.x * blockDim.x + threadIdx.x;
    int e = (int)(gid >> 5);
    int f = ((int)gid & 31) * 4;
    if (e >= E) return;
    int s = src[e];
    int d = dst[e];
    const float4 v = *(const float4*)(X + (size_t)s * HDIM + f);
    float* p = AGG + (size_t)d * HDIM + f;
    atomicAdd(p + 0, v.x);
    atomicAdd(p + 1, v.y);
    atomicAdd(p + 2, v.z);
    atomicAdd(p + 3, v.w);
}

// ---------------------------------------------------------------------------
// Fused GEMM:  OUT[rows x 128] = act( (A (+AGG)) @ W + bias )
// W is [128 x 128] row-major (fan_in, fan_out). rows % 16 == 0.
// Block = 256 threads (8 waves). Block computes 64 rows x 128 cols.
// Wave w owns columns [16w,16w+16) and 4 M-tiles (register-blocked), so each
// B fragment is reused by 4 V_WMMA_F32_16X16X4_F32 ops.
// ---------------------------------------------------------------------------
template <bool PREADD, bool RELU>
__global__ __launch_bounds__(256) void gemm128_kernel(
    const float* __restrict__ A, const float* __restrict__ AGG,
    const float* __restrict__ W, const float* __restrict__ bias,
    float* __restrict__ OUT, int rows) {
    __shared__ __align__(16) float lds_a[16 * MTILES][HDIM];

    const int tid  = threadIdx.x;
    const int wave = tid >> 5;
    const int lane = tid & 31;
    const int row0 = blockIdx.x * (16 * MTILES);

    // Stage the 64x128 A tile (optionally + AGG) in LDS: 2048 float4, 8/thread.
    // Out-of-range rows are zero-filled so all 4 M-tiles can compute uniformly.
    {
        const float4* a4 = (const float4*)(A + (size_t)row0 * HDIM);
        const float4* g4 = (const float4*)(AGG + (size_t)row0 * HDIM);
        float4* l4 = (float4*)&lds_a[0][0];
        #pragma unroll
        for (int i = tid; i < 16 * MTILES * (HDIM / 4); i += 256) {
            const int r = row0 + (i >> 5);     // 32 float4 per row
            float4 v = make_float4(0.f, 0.f, 0.f, 0.f);
            if (r < rows) {
                v = a4[i];
                if (PREADD) {
                    float4 g = g4[i];
                    v.x += g.x; v.y += g.y; v.z += g.z; v.w += g.w;
                }
            }
            l4[i] = v;
        }
    }
    __syncthreads();

    const int n0   = wave * 16;
    const int half = lane >> 4;   // 0: K-pair {0,1}; 1: K-pair {2,3}
    const int l16  = lane & 15;

    v8f c[MTILES];
    #pragma unroll
    for (int mt = 0; mt < MTILES; ++mt)
        c[mt] = (v8f){0.f, 0.f, 0.f, 0.f, 0.f, 0.f, 0.f, 0.f};

    #pragma unroll 4
    for (int k0 = 0; k0 < HDIM; k0 += 4) {
        const int ka = k0 + half * 2;
        v2f b;
        b.x = W[(size_t)ka * HDIM + n0 + l16];
        b.y = W[(size_t)(ka + 1) * HDIM + n0 + l16];
        #pragma unroll
        for (int mt = 0; mt < MTILES; ++mt) {
            v2f a;
            a.x = lds_a[mt * 16 + l16][ka];
            a.y = lds_a[mt * 16 + l16][ka + 1];
            c[mt] = __builtin_amdgcn_wmma_f32_16x16x4_f32(
                        /*neg_a=*/false, a, /*neg_b=*/false, b,
                        /*c_mod=*/(short)0, c[mt],
                        /*reuse_a=*/false, /*reuse_b=*/false);
        }
    }

    const float bv = bias[n0 + l16];
    #pragma unroll
    for (int mt = 0; mt < MTILES; ++mt) {
        const int tr0 = row0 + mt * 16;
        if (tr0 < rows) {                     // wave-uniform (rows % 16 == 0)
            #pragma unroll
            for (int v = 0; v < 8; ++v) {
                const int m = v + half * 8;
                float val = c[mt][v] + bv;
                if (RELU) val = fmaxf(val, 0.0f);
                OUT[(size_t)(tr0 + m) * HDIM + n0 + l16] = val;
            }
        }
    }
}

// ---------------------------------------------------------------------------
// Segment-sum pool: sums[batch[n]] += H[n], cnts[batch[n]] += 1
// ---------------------------------------------------------------------------
__global__ void pool_kernel(const float* __restrict__ H,
                            const int* __restrict__ batch,
                            float* __restrict__ sums,
                            float* __restrict__ cnts, int N) {
    long long gid = (long long)blockIdx.x * blockDim.x + threadIdx.x;
    int n = (int)(gid >> 5);
    int f = ((int)gid & 31) * 4;
    if (n >= N) return;
    int g = batch[n];
    const float4 v = *(const float4*)(H + (size_t)n * HDIM + f);
    float* p = sums + (size_t)g * HDIM + f;
    atomicAdd(p + 0, v.x);
    atomicAdd(p + 1, v.y);
    atomicAdd(p + 2, v.z);
    atomicAdd(p + 3, v.w);
    if (f == 0) atomicAdd(cnts + g, 1.0f);
}

// g = sums / max(cnts, 1)
__global__ void mean_kernel(const float* __restrict__ sums,
                            const float* __restrict__ cnts,
                            float* __restrict__ g) {
    int i = blockIdx.x * blockDim.x + threadIdx.x;
    if (i < 128 * HDIM) {
        float c = fmaxf(cnts[i >> 7], 1.0f);
        g[i] = sums[i] / c;
    }
}

// ---------------------------------------------------------------------------
// Dueling head epilogue, one wave32 per graph:
//   value = hv[r,:]@wv2 + bv2            (cross-lane reduced)
//   adv[a] = ha[r,:]@wa2[:,a] + ba2[a]   (lane a)
//   out[r,a] = value + adv[a] - mean(adv)
// ---------------------------------------------------------------------------
__global__ __launch_bounds__(32) void head_kernel(
    const float* __restrict__ hv, const float* __restrict__ ha,
    const float* __restrict__ wv2, const float* __restrict__ bv2,
    const float* __restrict__ wa2, const float* __restrict__ ba2,
    float* __restrict__ out) {
    const int r = blockIdx.x;
    const int a = threadIdx.x;   // 0..31

    float adv = ba2[a];
    #pragma unroll 4
    for (int k = 0; k < HDIM; ++k)
        adv += ha[(size_t)r * HDIM + k] * wa2[(size_t)k * 32 + a];

    float vp = 0.0f;
    #pragma unroll
    for (int k = a; k < HDIM; k += 32)
        vp += hv[(size_t)r * HDIM + k] * wv2[k];
    #pragma unroll
    for (int off = 16; off > 0; off >>= 1)
        vp += __shfl_down(vp, off, 32);
    float value = __shfl(vp, 0, 32) + bv2[0];

    float am = adv;
    #pragma unroll
    for (int off = 16; off > 0; off >>= 1)
        am += __shfl_down(am, off, 32);
    am = __shfl(am, 0, 32) * (1.0f / 32.0f);

    out[(size_t)r * 32 + a] = value + adv - am;
}

// ---------------------------------------------------------------------------
// Launch
// ---------------------------------------------------------------------------
extern "C" void kernel_launch(void* const* d_in, const int* in_sizes, int n_in,
                              void* d_out, int out_size, void* d_ws, size_t ws_size,
                              hipStream_t stream) {
    const float* x     = (const float*)d_in[0];
    const int*   ei    = (const int*)d_in[1];   // [2, E]
    const int*   batch = (const int*)d_in[2];
    const float* w1_1 = (const float*)d_in[3];
    const float* b1_1 = (const float*)d_in[4];
    const float* w1_2 = (const float*)d_in[5];
    const float* b1_2 = (const float*)d_in[6];
    const float* w2_1 = (const float*)d_in[7];
    const float* b2_1 = (const float*)d_in[8];
    const float* w2_2 = (const float*)d_in[9];
    const float* b2_2 = (const float*)d_in[10];
    const float* wv1  = (const float*)d_in[11];
    const float* bv1  = (const float*)d_in[12];
    const float* wv2  = (const float*)d_in[13];
    const float* bv2  = (const float*)d_in[14];
    const float* wa1  = (const float*)d_in[15];
    const float* ba1  = (const float*)d_in[16];
    const float* wa2  = (const float*)d_in[17];
    const float* ba2  = (const float*)d_in[18];

    const int E = in_sizes[1] / 2;
    const int N = in_sizes[2];
    const int* src = ei;
    const int* dst = ei + E;

    // Workspace layout (floats)
    float* ws   = (float*)d_ws;
    const size_t nodeF = (size_t)N * HDIM;
    float* agg  = ws;
    float* t    = agg + nodeF;
    float* l    = t + nodeF;
    float* sums = l + nodeF;
    float* cnts = sums + 128 * HDIM;   // contiguous after sums
    float* g    = cnts + 128;
    float* hv   = g + 128 * HDIM;
    float* ha   = hv + 128 * HDIM;

    const int zgrid = (int)((nodeF + 255) / 256);
    const int sgrid = (int)(((long long)E * 32 + 255) / 256);
    const int ggrid = (N + 16 * MTILES - 1) / (16 * MTILES);   // 782 blocks
    const int hgrid = (128 + 16 * MTILES - 1) / (16 * MTILES); // 2 blocks
    const int pgrid = (int)(((long long)N * 32 + 255) / 256);

    // ---- Layer 1: h = relu( relu((x+agg)@W1a+b) @ W1b + b ) ----
    zero_kernel<<<zgrid, 256, 0, stream>>>(agg, nodeF);
    scatter_add_kernel<<<sgrid, 256, 0, stream>>>(x, src, dst, agg, E);
    gemm128_kernel<true, true><<<ggrid, 256, 0, stream>>>(x, agg, w1_1, b1_1, t, N);
    gemm128_kernel<false, true><<<ggrid, 256, 0, stream>>>(t, t, w1_2, b1_2, l, N);

    // ---- Layer 2 ----
    zero_kernel<<<zgrid, 256, 0, stream>>>(agg, nodeF);
    scatter_add_kernel<<<sgrid, 256, 0, stream>>>(l, src, dst, agg, E);
    gemm128_kernel<true, true><<<ggrid, 256, 0, stream>>>(l, agg, w2_1, b2_1, t, N);
    gemm128_kernel<false, true><<<ggrid, 256, 0, stream>>>(t, t, w2_2, b2_2, l, N);

    // ---- Global mean pool ----
    zero_kernel<<<(128 * HDIM + 128 + 255) / 256, 256, 0, stream>>>(sums, 128 * HDIM + 128);
    pool_kernel<<<pgrid, 256, 0, stream>>>(l, batch, sums, cnts, N);
    mean_kernel<<<(128 * HDIM + 255) / 256, 256, 0, stream>>>(sums, cnts, g);

    // ---- Heads: hidden layers via WMMA GEMM (128 rows) ----
    gemm128_kernel<false, true><<<hgrid, 256, 0, stream>>>(g, g, wv1, bv1, hv, 128);
    gemm128_kernel<false, true><<<hgrid, 256, 0, stream>>>(g, g, wa1, ba1, ha, 128);

    // ---- Dueling combine ----
    head_kernel<<<128, 32, 0, stream>>>(hv, ha, wv2, bv2, wa2, ba2, (float*)d_out);
}